// AlgorithmReasoner_73572789781125
// MI455X (gfx1250) — compile-verified
//
#include <hip/hip_runtime.h>

// Sinkhorn scatter-log-softmax for MI455X (gfx1250, wave32).
// Memory-roofline kernel: ~12.3 GB streamed over 41 fused edge passes;
// atomics are native L2-side RMWs into 1.2MB node arrays (global_atomic_add_f32
// / global_atomic_max_i32, verified native via vmem histogram count).
// No dense-matmul structure => WMMA intentionally unused.

#define SK_NEG_INF (-1.0e6f)
#define SK_TB 256
static const int SK_NEG_INF_KEY = 0x807FFFFF; // fkey(-infinity)

// Monotone float<->signed-int key: signed-int order == float order.
__device__ __forceinline__ int fkey(float f) {
    int b = __float_as_int(f);
    return b ^ ((b >> 31) & 0x7FFFFFFF);
}
__device__ __forceinline__ float funkey(int k) {
    return __int_as_float(k ^ ((k >> 31) & 0x7FFFFFFF));
}

__global__ void sk_init_max(int* __restrict__ maxKey, int n) {
    int i = blockIdx.x * blockDim.x + threadIdx.x;
    if (i < n) maxKey[i] = SK_NEG_INF_KEY;
}

__global__ void sk_init_sum_max(float* __restrict__ sum, int* __restrict__ maxKey, int n) {
    int i = blockIdx.x * blockDim.x + threadIdx.x;
    if (i < n) {
        sum[i]    = 0.0f;
        maxKey[i] = SK_NEG_INF_KEY;
    }
}

// Node pass: fold per-node normalizer c[n] = max[n] + log(sum[n]) in place.
// 100K elements -> replaces 25.6M per-edge v_log with 100K, and halves the
// random-gather traffic of the finalize pass.
__global__ void sk_node_combine(float* __restrict__ sum, const int* __restrict__ maxKey, int n) {
    int i = blockIdx.x * blockDim.x + threadIdx.x;
    if (i < n) sum[i] = funkey(maxKey[i]) + __logf(sum[i]);
}

// Pass 0: y = y / T, mask self-loops to -1e6, write working copy to d_out,
// and seed the segment-max for the first (from_) softmax.
__global__ void sk_scale_mask_seed(const int* __restrict__ from_, const int* __restrict__ to_,
                                   const float* __restrict__ yin, const float* __restrict__ temp,
                                   float* __restrict__ yout, int* __restrict__ maxKey, int E) {
    const float invT = 1.0f / temp[0];
    int base = (blockIdx.x * blockDim.x + threadIdx.x) * 4;
    if (base + 3 < E) {
        int4   f  = *(const int4*)(from_ + base);
        int4   t  = *(const int4*)(to_   + base);
        float4 yv = *(const float4*)(yin + base);
        float o0 = (f.x == t.x) ? SK_NEG_INF : yv.x * invT;
        float o1 = (f.y == t.y) ? SK_NEG_INF : yv.y * invT;
        float o2 = (f.z == t.z) ? SK_NEG_INF : yv.z * invT;
        float o3 = (f.w == t.w) ? SK_NEG_INF : yv.w * invT;
        *(float4*)(yout + base) = make_float4(o0, o1, o2, o3);
        atomicMax(&maxKey[f.x], fkey(o0));
        atomicMax(&maxKey[f.y], fkey(o1));
        atomicMax(&maxKey[f.z], fkey(o2));
        atomicMax(&maxKey[f.w], fkey(o3));
    } else {
        for (int e = base; e < E; ++e) {
            int fi = from_[e], ti = to_[e];
            float o = (fi == ti) ? SK_NEG_INF : yin[e] * invT;
            yout[e] = o;
            atomicMax(&maxKey[fi], fkey(o));
        }
    }
}

// Pass B: sum[idx] += exp(y - max[idx])   (args <= 0, exp <= 1, no overflow)
__global__ void sk_pass_sum(const float* __restrict__ y, const int* __restrict__ idx,
                            const int* __restrict__ maxKey, float* __restrict__ sum, int E) {
    int base = (blockIdx.x * blockDim.x + threadIdx.x) * 4;
    __builtin_prefetch(y + base + 16384, 0, 1);
    __builtin_prefetch(idx + base + 16384, 0, 1);
    if (base + 3 < E) {
        float4 yv = *(const float4*)(y + base);
        int4   iv = *(const int4*)(idx + base);
        unsafeAtomicAdd(&sum[iv.x], __expf(yv.x - funkey(maxKey[iv.x])));
        unsafeAtomicAdd(&sum[iv.y], __expf(yv.y - funkey(maxKey[iv.y])));
        unsafeAtomicAdd(&sum[iv.z], __expf(yv.z - funkey(maxKey[iv.z])));
        unsafeAtomicAdd(&sum[iv.w], __expf(yv.w - funkey(maxKey[iv.w])));
    } else {
        for (int e = base; e < E; ++e) {
            int n = idx[e];
            unsafeAtomicAdd(&sum[n], __expf(y[e] - funkey(maxKey[n])));
        }
    }
}

// Pass C: y = y - c[idx]  (c = max + log(sum), pre-folded per node).
// Optionally seeds the NEXT softmax's segment-max (fused scatter saves one
// full edge pass per softmax).
__global__ void sk_pass_finalize(float* __restrict__ y, const int* __restrict__ idx,
                                 const float* __restrict__ c,
                                 const int* __restrict__ idxNext, int* __restrict__ maxKeyNext,
                                 int E) {
    int base = (blockIdx.x * blockDim.x + threadIdx.x) * 4;
    __builtin_prefetch(y + base + 16384, 0, 1);
    __builtin_prefetch(idx + base + 16384, 0, 1);
    if (base + 3 < E) {
        float4 yv = *(const float4*)(y + base);
        int4   iv = *(const int4*)(idx + base);
        float o0 = yv.x - c[iv.x];
        float o1 = yv.y - c[iv.y];
        float o2 = yv.z - c[iv.z];
        float o3 = yv.w - c[iv.w];
        *(float4*)(y + base) = make_float4(o0, o1, o2, o3);
        if (maxKeyNext) {
            int4 nv = *(const int4*)(idxNext + base);
            atomicMax(&maxKeyNext[nv.x], fkey(o0));
            atomicMax(&maxKeyNext[nv.y], fkey(o1));
            atomicMax(&maxKeyNext[nv.z], fkey(o2));
            atomicMax(&maxKeyNext[nv.w], fkey(o3));
        }
    } else {
        for (int e = base; e < E; ++e) {
            float o = y[e] - c[idx[e]];
            y[e] = o;
            if (maxKeyNext) atomicMax(&maxKeyNext[idxNext[e]], fkey(o));
        }
    }
}

extern "C" void kernel_launch(void* const* d_in, const int* in_sizes, int n_in,
                              void* d_out, int out_size, void* d_ws, size_t ws_size,
                              hipStream_t stream) {
    const int*   ei   = (const int*)d_in[0];    // [2, E]
    const float* yin  = (const float*)d_in[1];  // [E]
    const float* temp = (const float*)d_in[2];  // [1]
    const int E     = in_sizes[1];
    const int N     = 100000;   // static in reference (N_NODES)
    const int STEPS = 10;       // static in reference (steps)

    const int* from_ = ei;
    const int* to_   = ei + E;
    float* y = (float*)d_out;   // working buffer, iterated in place

    // Workspace layout: [maxA: N int][maxB: N int][sum/c: N float] = 1.2 MB
    int*   maxA = (int*)d_ws;
    int*   maxB = maxA + N;
    float* sum  = (float*)(maxB + N);
    (void)ws_size; (void)n_in; (void)out_size;

    const int nodeBlocks = (N + SK_TB - 1) / SK_TB;
    const int edgeBlocks = (((E + 3) / 4) + SK_TB - 1) / SK_TB;

    // Seed segment-max for step-0 "from" softmax while masking/scaling y.
    sk_init_max<<<nodeBlocks, SK_TB, 0, stream>>>(maxA, N);
    sk_scale_mask_seed<<<edgeBlocks, SK_TB, 0, stream>>>(from_, to_, yin, temp, y, maxA, E);

    for (int s = 0; s < STEPS; ++s) {
        // softmax over from_ : current max = maxA; feed maxB via to_
        sk_init_sum_max<<<nodeBlocks, SK_TB, 0, stream>>>(sum, maxB, N);
        sk_pass_sum<<<edgeBlocks, SK_TB, 0, stream>>>(y, from_, maxA, sum, E);
        sk_node_combine<<<nodeBlocks, SK_TB, 0, stream>>>(sum, maxA, N);
        sk_pass_finalize<<<edgeBlocks, SK_TB, 0, stream>>>(y, from_, sum, to_, maxB, E);

        // softmax over to_ : current max = maxB; feed maxA via from_ (unless last)
        sk_init_sum_max<<<nodeBlocks, SK_TB, 0, stream>>>(sum, maxA, N);
        sk_pass_sum<<<edgeBlocks, SK_TB, 0, stream>>>(y, to_, maxB, sum, E);
        sk_node_combine<<<nodeBlocks, SK_TB, 0, stream>>>(sum, maxB, N);
        int* feed = (s + 1 < STEPS) ? maxA : (int*)nullptr;
        sk_pass_finalize<<<edgeBlocks, SK_TB, 0, stream>>>(y, to_, sum, from_, feed, E);
    }
}